// SSIM_75290776699230
// MI455X (gfx1250) — compile-verified
//
#include <hip/hip_runtime.h>

// ---------------------------------------------------------------------------
// SSIM (16,3,512,512 fp32) for MI455X / gfx1250.
// Separable 11x11 Gaussian conv expressed as banded-matrix GEMMs on
// V_WMMA_F32_16X16X4_F32 (fp32 matrix rate). One wave32 per 16x16 out tile.
// ---------------------------------------------------------------------------

typedef float v2f __attribute__((ext_vector_type(2)));
typedef float v8f __attribute__((ext_vector_type(8)));

#define H_IMG   512
#define W_IMG   512
#define H_OUT   502          // 512 - 11 + 1
#define W_OUT   502
#define N_IMGS  48           // 16*3
#define TOTAL_OUT (48.0 * 502.0 * 502.0)

#define TS_X    29           // LDS stride for 32x28 input tiles (conflict-free)
#define TS_H    20           // LDS stride for 32x16 H tiles (conflict-free)

static __device__ __forceinline__ v8f wmma4(v2f a, v2f b, v8f c) {
    // D = A(16x4,f32) * B(4x16,f32) + C(16x16,f32); 8-arg VOP3P form.
    return __builtin_amdgcn_wmma_f32_16x16x4_f32(
        /*neg_a=*/false, a, /*neg_b=*/false, b,
        /*c_mod=*/(short)0, c, /*reuse_a=*/false, /*reuse_b=*/false);
}

__global__ __launch_bounds__(32) void ssim_zero_acc(double* __restrict__ ws) {
    ws[0] = 0.0;
}

__global__ __launch_bounds__(32) void ssim_tile_kernel(
    const float* __restrict__ x, const float* __restrict__ y,
    const float* __restrict__ win, double* __restrict__ acc)
{
    __shared__ float sx[32 * TS_X];
    __shared__ float sy[32 * TS_X];
    __shared__ float sH[5][32 * TS_H];
    __shared__ float sg[16];

    const int lane = threadIdx.x;                 // wave32
    const int ox0 = blockIdx.x * 16;
    const int oy0 = blockIdx.y * 16;
    const long base = (long)blockIdx.z * (H_IMG * W_IMG);

    // 1-D gaussian taps: g1[i] = sum_j window2d[i][j]  (window sums to 1)
    if (lane < 16) {
        float s = 0.0f;
        if (lane < 11) {
            #pragma unroll
            for (int j = 0; j < 11; ++j) s += win[lane * 11 + j];
        }
        sg[lane] = s;
    }

    // Fill 32x28 tiles (stride 29): rows>=26 / cols>=26 are ZERO (no NaN into
    // WMMA); zeroed region only feeds zero band coefficients anyway.
    for (int i = lane; i < 32 * 28; i += 32) {
        const int r = i / 28, c = i % 28;
        float vx = 0.0f, vy = 0.0f;
        if (r < 26 && c < 26) {
            int gr = oy0 + r; gr = (gr > H_IMG - 1) ? (H_IMG - 1) : gr;
            int gc = ox0 + c; gc = (gc > W_IMG - 1) ? (W_IMG - 1) : gc;
            const long gi = base + (long)gr * W_IMG + gc;
            vx = x[gi];
            vy = y[gi];
        }
        sx[r * TS_X + c] = vx;
        sy[r * TS_X + c] = vy;
    }
    __syncthreads();

    const int half   = (lane < 16) ? 0 : 2;       // A/B frag K sub-offset
    const int cm     = lane & 15;                 // column / M index in frag
    const int rowoff = (lane < 16) ? 0 : 8;       // C/D frag row offset

    // Banded coefficient fragments; identical for A (vertical, row-striped)
    // and B (horizontal, column-striped) because Gr = Gc^T.
    v2f coef[7];
    #pragma unroll
    for (int k = 0; k < 7; ++k) {
        const int j0 = 4 * k + half;
        const int d0 = j0 - cm, d1 = j0 + 1 - cm;
        coef[k].x = (d0 >= 0 && d0 <= 10) ? sg[d0] : 0.0f;
        coef[k].y = (d1 >= 0 && d1 <= 10) ? sg[d1] : 0.0f;
    }

    // ---- Horizontal pass: H[r][c] = sum_d g1[d] * In[r][c+d] -------------
    v8f h[5][2] = {};
    #pragma unroll
    for (int rh = 0; rh < 2; ++rh) {
        const int m = cm + rh * 16;               // data row for A frag
        #pragma unroll
        for (int k = 0; k < 7; ++k) {
            const int jj = 4 * k + half;
            v2f ax, ay;
            ax.x = sx[m * TS_X + jj];  ax.y = sx[m * TS_X + jj + 1];
            ay.x = sy[m * TS_X + jj];  ay.y = sy[m * TS_X + jj + 1];
            const v2f axx = ax * ax, ayy = ay * ay, axy = ax * ay;
            h[0][rh] = wmma4(ax,  coef[k], h[0][rh]);
            h[1][rh] = wmma4(ay,  coef[k], h[1][rh]);
            h[2][rh] = wmma4(axx, coef[k], h[2][rh]);
            h[3][rh] = wmma4(ayy, coef[k], h[3][rh]);
            h[4][rh] = wmma4(axy, coef[k], h[4][rh]);
        }
    }

    // Spill H (C/D layout) to LDS so it can be re-read in B layout.
    #pragma unroll
    for (int s = 0; s < 5; ++s)
        #pragma unroll
        for (int rh = 0; rh < 2; ++rh)
            #pragma unroll
            for (int v = 0; v < 8; ++v)
                sH[s][(rh * 16 + v + rowoff) * TS_H + cm] = h[s][rh][v];
    __syncthreads();

    // ---- Vertical pass: Out[m][n] = sum_d g1[d] * H[m+d][n] --------------
    v8f o[5] = {};
    #pragma unroll
    for (int k = 0; k < 7; ++k) {
        const int jj = 4 * k + half;
        #pragma unroll
        for (int s = 0; s < 5; ++s) {
            v2f b;
            b.x = sH[s][jj * TS_H + cm];
            b.y = sH[s][(jj + 1) * TS_H + cm];
            o[s] = wmma4(coef[k], b, o[s]);
        }
    }

    // ---- SSIM map + wave reduction ---------------------------------------
    const float c1 = 1.0e-4f;                     // (0.01*1)^2
    const float c2 = 9.0e-4f;                     // (0.03*1)^2
    float lsum = 0.0f;
    #pragma unroll
    for (int v = 0; v < 8; ++v) {
        const int m  = v + rowoff;
        const int oy = oy0 + m;
        const int ox = ox0 + cm;
        const float mux = o[0][v], muy = o[1][v];
        const float sxx = o[2][v] - mux * mux;
        const float syy = o[3][v] - muy * muy;
        const float sxy = o[4][v] - mux * muy;
        const float cs  = (2.0f * sxy + c2) / (sxx + syy + c2);
        const float lum = (2.0f * mux * muy + c1) /
                          (mux * mux + muy * muy + c1);
        lsum += (oy < H_OUT && ox < W_OUT) ? (lum * cs) : 0.0f;
    }
    #pragma unroll
    for (int off = 16; off > 0; off >>= 1)
        lsum += __shfl_xor(lsum, off, 32);
    if (lane == 0) atomicAdd(acc, (double)lsum);
}

__global__ __launch_bounds__(32) void ssim_finalize(
    const double* __restrict__ acc, float* __restrict__ out)
{
    out[0] = (float)(acc[0] / TOTAL_OUT);
}

extern "C" void kernel_launch(void* const* d_in, const int* in_sizes, int n_in,
                              void* d_out, int out_size, void* d_ws, size_t ws_size,
                              hipStream_t stream)
{
    const float* x   = (const float*)d_in[0];   // (16,3,512,512)
    const float* y   = (const float*)d_in[1];   // (16,3,512,512)
    const float* win = (const float*)d_in[2];   // (11,11)
    float* out = (float*)d_out;
    double* acc = (double*)d_ws;                // fp64 accumulator in scratch

    ssim_zero_acc<<<1, 32, 0, stream>>>(acc);

    dim3 grid((W_OUT + 15) / 16, (H_OUT + 15) / 16, N_IMGS);  // 32 x 32 x 48
    ssim_tile_kernel<<<grid, 32, 0, stream>>>(x, y, win, acc);

    ssim_finalize<<<1, 32, 0, stream>>>(acc, out);
}